// GCNe_48593259987018
// MI455X (gfx1250) — compile-verified
//
#include <hip/hip_runtime.h>
#include <hip/hip_bf16.h>

typedef __attribute__((ext_vector_type(2))) float v2f;
typedef __attribute__((ext_vector_type(8))) float v8f;

// ---------------------------------------------------------------------------
// Kernel 1: edge gather + scale + scatter-add.
// One wave32 per edge: 64 features = 2 floats per lane -> two coalesced
// 128B lines read from x[src], two coalesced atomic-add bursts into agg[dst].
// x / agg are 25.6 MB -> L2 resident on the 192 MB L2; atomics hit L2.
// ---------------------------------------------------------------------------
__global__ __launch_bounds__(256)
void gcn_scatter_kernel(const float* __restrict__ x,
                        const int* __restrict__ src,
                        const int* __restrict__ dst,
                        const float* __restrict__ ew,
                        float* __restrict__ agg,
                        int E) {
    int lane = threadIdx.x & 31;
    int edge = (int)((blockIdx.x * (unsigned)blockDim.x + threadIdx.x) >> 5);
    if (edge >= E) return;
    int s = src[edge];
    int d = dst[edge];
    float w = ew[edge];
    float v0 = x[s * 64 + lane] * w;
    float v1 = x[s * 64 + 32 + lane] * w;
    atomicAdd(&agg[d * 64 + lane], v0);
    atomicAdd(&agg[d * 64 + 32 + lane], v1);
}

// ---------------------------------------------------------------------------
// Kernel 2: fused dual GEMM + bias + ReLU using V_WMMA_F32_16X16X4_F32.
//   out = relu(agg @ Wr + x @ Ws + b), all [.,64] x [64,64].
// Block = 256 threads = 8 waves; each wave computes a 16-row x 64-col strip.
//
// Weights are staged into LDS PRE-SWIZZLED into WMMA B-fragment order:
// pair index p = ((kk*4 + t)*32 + lane) holds this lane's contiguous
// (W[ka][col], W[ka+1][col]) pair (ka = 4*kk + 2*half, col = 16*t + l16),
// so the inner loop is one ds_load_b64 per fragment -> adjacent even-aligned
// VGPR pair, zero repacking movs, conflict-free across 64 banks.
//
// Fragment layouts per CDNA5 ISA 7.12.2:
//   A 16x4 f32 : lanes 0-15 {K0,K1}, lanes 16-31 {K2,K3}   (2 VGPRs)
//   B 4x16 f32 : mirrored (VGPR0 = K(2*half), VGPR1 = K(2*half+1))
//   C/D 16x16  : VGPR j -> row j (lanes 0-15) / row j+8 (lanes 16-31)
// ---------------------------------------------------------------------------
__global__ __launch_bounds__(256)
void gcn_gemm_kernel(const float* __restrict__ agg,
                     const float* __restrict__ x,
                     const float* __restrict__ Wr,
                     const float* __restrict__ Ws,
                     const float* __restrict__ bias,
                     float* __restrict__ out,
                     int N) {
    __shared__ float sWr[64 * 64];   // fragment-ordered, 16 KB
    __shared__ float sWs[64 * 64];   // fragment-ordered, 16 KB

    // Stage weights in fragment order (one-time, 32 KB per block).
    for (int p = threadIdx.x; p < 2048; p += 256) {
        const int f   = p >> 5;          // fragment id: kk*4 + t  (0..63)
        const int l   = p & 31;          // lane within fragment
        const int kk  = f >> 2;
        const int t   = f & 3;
        const int hlf = l >> 4;
        const int l16 = l & 15;
        const int ka  = kk * 4 + 2 * hlf;
        const int col = t * 16 + l16;
        sWr[2 * p]     = Wr[ka * 64 + col];
        sWr[2 * p + 1] = Wr[(ka + 1) * 64 + col];
        sWs[2 * p]     = Ws[ka * 64 + col];
        sWs[2 * p + 1] = Ws[(ka + 1) * 64 + col];
    }
    __syncthreads();

    const int wave = threadIdx.x >> 5;
    const int lane = threadIdx.x & 31;
    const int half = lane >> 4;    // 0: lanes 0-15, 1: lanes 16-31
    const int l16  = lane & 15;

    const int m0 = blockIdx.x * 128 + wave * 16;   // first node row of strip
    int arow = m0 + l16;
    if (arow >= N) arow = N - 1;                   // clamp (EXEC stays full)
    const float* __restrict__ aggRow = agg + (long)arow * 64;
    const float* __restrict__ xRow   = x   + (long)arow * 64;

    const v2f* __restrict__ fWr = (const v2f*)sWr;
    const v2f* __restrict__ fWs = (const v2f*)sWs;

    v8f c0 = {}, c1 = {}, c2 = {}, c3 = {};

    #pragma unroll
    for (int kk = 0; kk < 16; ++kk) {
        const int ka = kk * 4 + 2 * half;          // this lane's K pair
        v2f aA = *(const v2f*)(aggRow + ka);       // global_load_b64
        v2f aX = *(const v2f*)(xRow + ka);         // global_load_b64

        const int base = kk * 128 + lane;          // (kk*4 + t)*32 + lane
        v2f bR0 = fWr[base +  0];                  // ds_load_b64 each
        v2f bR1 = fWr[base + 32];
        v2f bR2 = fWr[base + 64];
        v2f bR3 = fWr[base + 96];
        v2f bS0 = fWs[base +  0];
        v2f bS1 = fWs[base + 32];
        v2f bS2 = fWs[base + 64];
        v2f bS3 = fWs[base + 96];

        c0 = __builtin_amdgcn_wmma_f32_16x16x4_f32(false, aA, false, bR0, (short)0, c0, false, false);
        c1 = __builtin_amdgcn_wmma_f32_16x16x4_f32(false, aA, false, bR1, (short)0, c1, false, false);
        c2 = __builtin_amdgcn_wmma_f32_16x16x4_f32(false, aA, false, bR2, (short)0, c2, false, false);
        c3 = __builtin_amdgcn_wmma_f32_16x16x4_f32(false, aA, false, bR3, (short)0, c3, false, false);
        c0 = __builtin_amdgcn_wmma_f32_16x16x4_f32(false, aX, false, bS0, (short)0, c0, false, false);
        c1 = __builtin_amdgcn_wmma_f32_16x16x4_f32(false, aX, false, bS1, (short)0, c1, false, false);
        c2 = __builtin_amdgcn_wmma_f32_16x16x4_f32(false, aX, false, bS2, (short)0, c2, false, false);
        c3 = __builtin_amdgcn_wmma_f32_16x16x4_f32(false, aX, false, bS3, (short)0, c3, false, false);
    }

    const float b0 = bias[0  + l16];
    const float b1 = bias[16 + l16];
    const float b2 = bias[32 + l16];
    const float b3 = bias[48 + l16];

    #pragma unroll
    for (int j = 0; j < 8; ++j) {
        const int row = m0 + j + 8 * half;         // D layout: VGPR j -> row j / j+8
        if (row < N) {
            float* o = out + (long)row * 64;
            o[0  + l16] = fmaxf(c0[j] + b0, 0.0f);
            o[16 + l16] = fmaxf(c1[j] + b1, 0.0f);
            o[32 + l16] = fmaxf(c2[j] + b2, 0.0f);
            o[48 + l16] = fmaxf(c3[j] + b3, 0.0f);
        }
    }
}

// ---------------------------------------------------------------------------
// Kernel 3: global mean-pool accumulation (sums + counts) via atomics.
// ---------------------------------------------------------------------------
__global__ __launch_bounds__(256)
void gcn_pool_kernel(const float* __restrict__ h,
                     const int* __restrict__ batch,
                     float* __restrict__ sums,
                     float* __restrict__ cnts,
                     int N) {
    long t = (long)blockIdx.x * blockDim.x + threadIdx.x;
    if (t >= (long)N * 64) return;
    int n = (int)(t >> 6);
    int f = (int)(t & 63);
    int g = batch[n];
    atomicAdd(&sums[g * 64 + f], h[t]);
    if (f == 0) atomicAdd(&cnts[g], 1.0f);
}

// ---------------------------------------------------------------------------
// Kernel 4: mean + head GEMM [256,64] x [64,10] + bias. Tiny; plain VALU.
// ---------------------------------------------------------------------------
__global__ __launch_bounds__(256)
void gcn_head_kernel(const float* __restrict__ sums,
                     const float* __restrict__ cnts,
                     const float* __restrict__ Wl,
                     const float* __restrict__ bl,
                     float* __restrict__ out,
                     int G, int C) {
    int t = blockIdx.x * blockDim.x + threadIdx.x;
    if (t >= G * C) return;
    int g = t / C;
    int c = t % C;
    float inv = 1.0f / fmaxf(cnts[g], 1.0f);
    float acc = bl[c];
    #pragma unroll 8
    for (int hh = 0; hh < 64; ++hh)
        acc += (sums[g * 64 + hh] * inv) * Wl[hh * C + c];
    out[t] = acc;
}

// ---------------------------------------------------------------------------
// Host-side orchestration (graph-capture safe: only kernel launches and
// hipMemsetAsync on `stream`; scratch comes from d_ws).
// ---------------------------------------------------------------------------
extern "C" void kernel_launch(void* const* d_in, const int* in_sizes, int n_in,
                              void* d_out, int out_size, void* d_ws, size_t ws_size,
                              hipStream_t stream) {
    const float* x    = (const float*)d_in[0];
    const int*   ei   = (const int*)  d_in[1];   // [2, E]
    const float* ew   = (const float*)d_in[2];
    const int*   bat  = (const int*)  d_in[3];
    const float* W1r  = (const float*)d_in[4];
    const float* W1s  = (const float*)d_in[5];
    const float* b1   = (const float*)d_in[6];
    const float* W2r  = (const float*)d_in[7];
    const float* W2s  = (const float*)d_in[8];
    const float* b2   = (const float*)d_in[9];
    const float* W3r  = (const float*)d_in[10];
    const float* W3s  = (const float*)d_in[11];
    const float* b3   = (const float*)d_in[12];
    const float* Wl   = (const float*)d_in[13];
    const float* bl   = (const float*)d_in[14];

    const int N = in_sizes[0] / 64;
    const int E = in_sizes[1] / 2;
    const int G = 256;
    const int C = 10;

    const int* src = ei;
    const int* dst = ei + E;

    // Workspace layout
    float* agg  = (float*)d_ws;                   // [N,64]
    float* hA   = agg  + (size_t)N * 64;          // [N,64]
    float* hB   = hA   + (size_t)N * 64;          // [N,64]
    float* sums = hB   + (size_t)N * 64;          // [G,64]
    float* cnts = sums + (size_t)G * 64;          // [G]

    const int scatterBlocks = (E + 7) / 8;        // 8 waves (edges) per 256-thr block
    const int gemmBlocks    = (N + 127) / 128;    // 8 waves x 16 rows per block
    const int poolBlocks    = (int)(((long)N * 64 + 255) / 256);
    const int headBlocks    = (G * C + 255) / 256;

    // ----- Layer 1: x -> hA -----
    hipMemsetAsync(agg, 0, (size_t)N * 64 * sizeof(float), stream);
    gcn_scatter_kernel<<<scatterBlocks, 256, 0, stream>>>(x, src, dst, ew, agg, E);
    gcn_gemm_kernel<<<gemmBlocks, 256, 0, stream>>>(agg, x, W1r, W1s, b1, hA, N);

    // ----- Layer 2: hA -> hB -----
    hipMemsetAsync(agg, 0, (size_t)N * 64 * sizeof(float), stream);
    gcn_scatter_kernel<<<scatterBlocks, 256, 0, stream>>>(hA, src, dst, ew, agg, E);
    gcn_gemm_kernel<<<gemmBlocks, 256, 0, stream>>>(agg, hA, W2r, W2s, b2, hB, N);

    // ----- Layer 3: hB -> hA -----
    hipMemsetAsync(agg, 0, (size_t)N * 64 * sizeof(float), stream);
    gcn_scatter_kernel<<<scatterBlocks, 256, 0, stream>>>(hB, src, dst, ew, agg, E);
    gcn_gemm_kernel<<<gemmBlocks, 256, 0, stream>>>(agg, hB, W3r, W3s, b3, hA, N);

    // ----- Pool + head -----
    hipMemsetAsync(sums, 0, ((size_t)G * 64 + G) * sizeof(float), stream);
    gcn_pool_kernel<<<poolBlocks, 256, 0, stream>>>(hA, bat, sums, cnts, N);
    gcn_head_kernel<<<headBlocks, 256, 0, stream>>>(sums, cnts, Wl, bl, (float*)d_out, G, C);
}